// VQVAE_66649302499910
// MI455X (gfx1250) — compile-verified
//
#include <hip/hip_runtime.h>

typedef __attribute__((ext_vector_type(2))) float v2f;
typedef __attribute__((ext_vector_type(8))) float v8f;

#define NTOK   16384      // B*H*W = 4*64*64
#define KCODES 8192
#define CHUNK  2048       // codes staged in LDS per phase (32 KB)
#define HWSZ   4096       // H*W
#define LOSS_SCALE (1.25f / 65536.0f)   // (1+BETA)/(B*N*C)

__global__ void vq_init_kernel(float* __restrict__ out) {
    out[NTOK * 4] = 0.0f;   // zero the loss accumulator slot
}

__global__ __launch_bounds__(256) void vqvae_wmma_kernel(
    const float* __restrict__ z,    // [4,4,64,64]
    const float* __restrict__ wq,   // [4,4] row-major [out_c, in_c]
    const float* __restrict__ bq,   // [4]
    const float* __restrict__ cb,   // [8192,4]
    float* __restrict__ out)        // [65536] quant_st + [1] loss
{
    __shared__ float4 lcb[CHUNK];

    const int lane = threadIdx.x & 31;
    const int wav  = threadIdx.x >> 5;          // 8 waves / block
    const int col  = lane & 15;                 // code column within tile
    const bool hi  = lane >= 16;
    const int t0   = (blockIdx.x * 8 + wav) * 16;  // 16 tokens per wave

    // ---- pre-quant weights (uniform -> scalar loads) ----
    float W[16], Bb[4];
#pragma unroll
    for (int i = 0; i < 16; ++i) W[i] = wq[i];
#pragma unroll
    for (int i = 0; i < 4; ++i)  Bb[i] = bq[i];

    // ---- load token, apply 1x1 pre-quant conv ----
    // lane L computes x for token t0 + (L&15); lanes 16-31 mirror for A's K=2,3 half
    const int t  = t0 + col;
    const int bi = t >> 12;               // batch
    const int hw = t & (HWSZ - 1);
    const float* zp = z + (bi << 14) + hw;      // b*C*HW
    const float z0 = zp[0], z1 = zp[HWSZ], z2 = zp[2 * HWSZ], z3 = zp[3 * HWSZ];

    const float x0 = fmaf(W[0],  z0, fmaf(W[1],  z1, fmaf(W[2],  z2, fmaf(W[3],  z3, Bb[0]))));
    const float x1 = fmaf(W[4],  z0, fmaf(W[5],  z1, fmaf(W[6],  z2, fmaf(W[7],  z3, Bb[1]))));
    const float x2 = fmaf(W[8],  z0, fmaf(W[9],  z1, fmaf(W[10], z2, fmaf(W[11], z3, Bb[2]))));
    const float x3 = fmaf(W[12], z0, fmaf(W[13], z1, fmaf(W[14], z2, fmaf(W[15], z3, Bb[3]))));

    // A matrix (16x4 f32): lanes 0-15 hold {K=0,K=1} of row M=lane,
    // lanes 16-31 hold {K=2,K=3}. Fold the -2 of the distance formula in here.
    v2f A;
    A.x = -2.0f * (hi ? x2 : x0);
    A.y = -2.0f * (hi ? x3 : x1);

    // running min of (dist - ||x||^2) and argmin, per D-tile register slot
    float best[8];
    int   bkid[8];
#pragma unroll
    for (int r = 0; r < 8; ++r) { best[r] = 3.4e38f; bkid[r] = 0; }

    const float4* cb4 = (const float4*)cb;

    for (int base = 0; base < KCODES; base += CHUNK) {
        __syncthreads();                         // all waves done reading prev chunk
        for (int i = 0; i < CHUNK / 256; ++i)    // cooperative stage: 8 x b128 per thread
            lcb[threadIdx.x + i * 256] = cb4[base + threadIdx.x + i * 256];
        __syncthreads();

#pragma unroll 2
        for (int kk = 0; kk < CHUNK; kk += 16) {
            const float4 c = lcb[kk + col];      // lanes L and L+16 broadcast-read same code
            const float cn = fmaf(c.x, c.x, fmaf(c.y, c.y, fmaf(c.z, c.z, c.w * c.w)));

            // B matrix (4x16 f32): lane N<16 holds {K=0,K=1} of column N,
            // lane N+16 holds {K=2,K=3} of column N.
            v2f Bm;
            Bm.x = hi ? c.z : c.x;
            Bm.y = hi ? c.w : c.y;

            // C = ||c||^2 broadcast per column -> D = ||c||^2 - 2 x.c  (argmin-equivalent)
            v8f Cm = {cn, cn, cn, cn, cn, cn, cn, cn};
            v8f D = __builtin_amdgcn_wmma_f32_16x16x4_f32(
                false, A, false, Bm, (short)0, Cm, false, false);

            const int kidx = base + kk + col;
#pragma unroll
            for (int r = 0; r < 8; ++r) {
                if (D[r] < best[r]) { best[r] = D[r]; bkid[r] = kidx; }
            }
        }
    }

    // ---- argmin reduce across the 16 code-column lanes of each half ----
#pragma unroll
    for (int r = 0; r < 8; ++r) {
        float bv = best[r]; int bk = bkid[r];
#pragma unroll
        for (int m = 1; m < 16; m <<= 1) {
            const float ov = __shfl_xor(bv, m, 32);
            const int   ok = __shfl_xor(bk, m, 32);
            if (ov < bv || (ov == bv && ok < bk)) { bv = ov; bk = ok; }
        }
        best[r] = bv; bkid[r] = bk;
    }

    // map token m -> lane m (m = 0..15): rows 0-7 live in lanes 0-15 (reg r = m),
    // rows 8-15 live in lanes 16-31 (reg r = m-8) -> pull across with xor-16 shuffle
    int myk = 0;
#pragma unroll
    for (int r = 0; r < 8; ++r) {
        const int khi = __shfl_xor(bkid[r], 16, 32);
        if (col == r)     myk = bkid[r];
        if (col == r + 8) myk = khi;
    }

    // ---- gather, straight-through output, loss partial ----
    float lpart = 0.0f;
    if (!hi) {
        const float4 q = cb4[myk];
        const float d0 = q.x - x0, d1 = q.y - x1, d2 = q.z - x2, d3 = q.w - x3;
        // straight-through: x + (quant - x), same rounding order as reference
        const float q0 = x0 + d0, q1 = x1 + d1, q2 = x2 + d2, q3 = x3 + d3;

        const int tt = t0 + lane;
        const int b2 = tt >> 12, hw2 = tt & (HWSZ - 1);
        float* op = out + (b2 << 14) + hw2;     // [B,C,H,W] scatter
        op[0]        = q0;
        op[HWSZ]     = q1;
        op[2 * HWSZ] = q2;
        op[3 * HWSZ] = q3;

        lpart = fmaf(d0, d0, fmaf(d1, d1, fmaf(d2, d2, d3 * d3)));
    }
#pragma unroll
    for (int m = 1; m < 32; m <<= 1)
        lpart += __shfl_xor(lpart, m, 32);
    if (lane == 0)
        atomicAdd(out + NTOK * 4, lpart * LOSS_SCALE);
}

extern "C" void kernel_launch(void* const* d_in, const int* in_sizes, int n_in,
                              void* d_out, int out_size, void* d_ws, size_t ws_size,
                              hipStream_t stream) {
    const float* z  = (const float*)d_in[0];   // 65536
    const float* wq = (const float*)d_in[1];   // 16
    const float* bq = (const float*)d_in[2];   // 4
    const float* cb = (const float*)d_in[3];   // 32768
    float* out = (float*)d_out;                // 65537

    vq_init_kernel<<<1, 1, 0, stream>>>(out);
    // 16384 tokens / (8 waves * 16 tokens) = 128 blocks of 256 threads
    vqvae_wmma_kernel<<<128, 256, 0, stream>>>(z, wq, bq, cb, out);
}